// GNNEncoder_11871289606581
// MI455X (gfx1250) — compile-verified
//
#include <hip/hip_runtime.h>

typedef float v2f __attribute__((ext_vector_type(2)));
typedef float v8f __attribute__((ext_vector_type(8)));

#define GCN_N 100000
#define GCN_E 1600000

__device__ __forceinline__ void atomic_add_f32(float* p, float v) {
    // lowers to global_atomic_add_f32 (hardware FP atomic, L2-resident here)
    unsafeAtomicAdd(p, v);
}

__global__ void deg_init_kernel(float* deg, int n) {
    int i = blockIdx.x * blockDim.x + threadIdx.x;
    if (i < n) deg[i] = 1.0f;  // self loop contributes 1 to degree
}

__global__ void deg_count_kernel(const int* __restrict__ dst, float* deg, int e) {
    int i = blockIdx.x * blockDim.x + threadIdx.x;
    if (i < e) atomic_add_f32(&deg[dst[i]], 1.0f);
}

__global__ void dinv_kernel(float* d, int n) {
    int i = blockIdx.x * blockDim.x + threadIdx.x;
    if (i < n) d[i] = rsqrtf(d[i]);  // deg >= 1 always (self loop)
}

// H[M,NOUT] = act(X[M,KDIM]) @ W[KDIM,NOUT] with V_WMMA_F32_16X16X4_F32.
// One wave owns exactly one 16x16 output tile: single v8f accumulator,
// ~20 live VGPRs -> no spills. All shapes compile-time.
template <int KDIM, int NOUT, bool RELU>
__global__ void gemm_wmma_kernel(const float* __restrict__ X,
                                 const float* __restrict__ W,
                                 float* __restrict__ H, int M) {
    constexpr int NT  = NOUT / 16;   // n-tiles per row block
    constexpr int LDW = NOUT + 8;    // pad: lane-halves hit disjoint LDS banks
    __shared__ float Ws[KDIM * LDW];
    for (int i = threadIdx.x; i < KDIM * NOUT; i += blockDim.x) {
        int k = i / NOUT, n = i - k * NOUT;  // NOUT is pow2 -> shifts
        Ws[k * LDW + n] = W[i];
    }
    __syncthreads();

    const int wid  = (blockIdx.x * blockDim.x + threadIdx.x) >> 5;
    const int lane = threadIdx.x & 31;
    const int m0 = (wid / NT) * 16;
    const int n0 = (wid % NT) * 16;
    if (m0 >= M) return;  // wave-uniform; surviving waves keep EXEC all-1s

    const int half = lane >> 4;   // 0: K pair {0,1}; 1: K pair {2,3}
    const int l16  = lane & 15;
    const int koff = half * 2;
    // A fragment: 16x4 f32 layout — row = l16, VGPR0=K(koff), VGPR1=K(koff+1)
    const float* xrow = X + (size_t)(m0 + l16) * KDIM + koff;
    // B fragment: 4x16 f32 layout — col = l16, VGPR0=K(koff), VGPR1=K(koff+1)
    const float* wcol = Ws + koff * LDW + n0 + l16;

    v8f acc = {};
#pragma unroll 4
    for (int k0 = 0; k0 < KDIM; k0 += 4) {
        float a0 = xrow[k0];
        float a1 = xrow[k0 + 1];
        if (RELU) { a0 = fmaxf(a0, 0.0f); a1 = fmaxf(a1, 0.0f); }
        v2f a; a.x = a0; a.y = a1;
        v2f b; b.x = wcol[k0 * LDW]; b.y = wcol[k0 * LDW + LDW];
        acc = __builtin_amdgcn_wmma_f32_16x16x4_f32(
            false, a, false, b, (short)0, acc, false, false);
    }
    // D layout: VGPR v -> row m0 + v + 8*half, col = n0 + l16
    float* orow = H + (size_t)(m0 + half * 8) * NOUT + n0 + l16;
#pragma unroll
    for (int v = 0; v < 8; ++v)
        orow[(size_t)v * NOUT] = acc[v];
}

// A[i,:] = H[i,:] * dinv[i]^2 + b   (self-loop term + bias; fully overwrites A)
__global__ void agg_init_kernel(const float* __restrict__ H,
                                const float* __restrict__ dinv,
                                const float* __restrict__ bias,
                                float* __restrict__ A, int n, int fsh) {
    int t = blockIdx.x * blockDim.x + threadIdx.x;
    if (t >= (n << fsh)) return;
    int i = t >> fsh;
    int c = (t & ((1 << fsh) - 1)) << 2;
    int F = 4 << fsh;
    float d = dinv[i];
    float s = d * d;
    float4 h = *(const float4*)(H + (size_t)i * F + c);
    float4 o;
    o.x = h.x * s + bias[c + 0];
    o.y = h.y * s + bias[c + 1];
    o.z = h.z * s + bias[c + 2];
    o.w = h.w * s + bias[c + 3];
    *(float4*)(A + (size_t)i * F + c) = o;
}

// A[dst,:] += H[src,:] * dinv[src]*dinv[dst]; one thread per (edge, 4 features)
__global__ void agg_edge_kernel(const int* __restrict__ src,
                                const int* __restrict__ dst,
                                const float* __restrict__ dinv,
                                const float* __restrict__ H,
                                float* __restrict__ A, int E, int fsh) {
    int t = blockIdx.x * blockDim.x + threadIdx.x;
    if (t >= (E << fsh)) return;
    int e = t >> fsh;
    int c = (t & ((1 << fsh) - 1)) << 2;
    int F = 4 << fsh;
    int s = src[e], d = dst[e];
    float nrm = dinv[s] * dinv[d];
    float4 h = *(const float4*)(H + (size_t)s * F + c);
    float* out = A + (size_t)d * F + c;
    atomic_add_f32(out + 0, h.x * nrm);
    atomic_add_f32(out + 1, h.y * nrm);
    atomic_add_f32(out + 2, h.z * nrm);
    atomic_add_f32(out + 3, h.w * nrm);
}

extern "C" void kernel_launch(void* const* d_in, const int* in_sizes, int n_in,
                              void* d_out, int out_size, void* d_ws, size_t ws_size,
                              hipStream_t stream) {
    (void)in_sizes; (void)n_in; (void)out_size;
    const float* x  = (const float*)d_in[0];
    const int*   ei = (const int*)d_in[1];
    const float* W1 = (const float*)d_in[2];
    const float* b1 = (const float*)d_in[3];
    const float* W2 = (const float*)d_in[4];
    const float* b2 = (const float*)d_in[5];
    const float* W3 = (const float*)d_in[6];
    const float* b3 = (const float*)d_in[7];
    float* out = (float*)d_out;

    const int N = GCN_N, E = GCN_E;
    const int* srcI = ei;
    const int* dstI = ei + E;

    float* dinv = (float*)d_ws;
    float* Hbuf = dinv + N;                    // N x 64 GEMM output
    float* Abuf = Hbuf + (size_t)N * 64;       // N x 64 aggregated output
    size_t need = ((size_t)N + 2 * (size_t)N * 64) * sizeof(float);
    if (ws_size < need) return;

    const int TB = 256;
    const int mtiles = N / 16;                         // 6250, exact
    const int wavesA = mtiles * 4;                     // NOUT=64 -> 4 n-tiles
    const int wavesB = mtiles * 2;                     // NOUT=32 -> 2 n-tiles
    const int gbA = (wavesA + 7) / 8;                  // 8 waves / 256-thr block
    const int gbB = (wavesB + 7) / 8;

    // Degree / normalization (recomputed each call: graph-replay safe)
    deg_init_kernel <<<(N + TB - 1) / TB, TB, 0, stream>>>(dinv, N);
    deg_count_kernel<<<(E + TB - 1) / TB, TB, 0, stream>>>(dstI, dinv, E);
    dinv_kernel     <<<(N + TB - 1) / TB, TB, 0, stream>>>(dinv, N);

    // Layer 1: H1 = x @ W1 ; A1 = scatter(H1) + b1  (ReLU fused into next GEMM read)
    gemm_wmma_kernel<128, 64, false><<<gbA, TB, 0, stream>>>(x, W1, Hbuf, N);
    agg_init_kernel<<<((N << 4) + TB - 1) / TB, TB, 0, stream>>>(Hbuf, dinv, b1, Abuf, N, 4);
    agg_edge_kernel<<<((E << 4) + TB - 1) / TB, TB, 0, stream>>>(srcI, dstI, dinv, Hbuf, Abuf, E, 4);

    // Layer 2: H2 = relu(A1) @ W2 ; A2 = scatter(H2) + b2
    gemm_wmma_kernel<64, 64, true><<<gbA, TB, 0, stream>>>(Abuf, W2, Hbuf, N);
    agg_init_kernel<<<((N << 4) + TB - 1) / TB, TB, 0, stream>>>(Hbuf, dinv, b2, Abuf, N, 4);
    agg_edge_kernel<<<((E << 4) + TB - 1) / TB, TB, 0, stream>>>(srcI, dstI, dinv, Hbuf, Abuf, E, 4);

    // Layer 3: H3 = relu(A2) @ W3 ; out = scatter(H3) + b3  (no ReLU)
    gemm_wmma_kernel<64, 32, true><<<gbB, TB, 0, stream>>>(Abuf, W3, Hbuf, N);
    agg_init_kernel<<<((N << 3) + TB - 1) / TB, TB, 0, stream>>>(Hbuf, dinv, b3, out, N, 3);
    agg_edge_kernel<<<((E << 3) + TB - 1) / TB, TB, 0, stream>>>(srcI, dstI, dinv, Hbuf, out, E, 3);
}